// ScalarMapping_26912265077083
// MI455X (gfx1250) — compile-verified
//
#include <hip/hip_runtime.h>
#include <stdint.h>

typedef __attribute__((ext_vector_type(2))) float v2f;
typedef __attribute__((ext_vector_type(4))) float v4f;
typedef __attribute__((ext_vector_type(8))) float v8f;

typedef int v4i __attribute__((vector_size(16)));
typedef __attribute__((address_space(1))) v4i g_v4i;   // global-AS int4
typedef __attribute__((address_space(3))) v4i l_v4i;   // LDS-AS int4

constexpr int kB = 256;
constexpr int kC = 512;
constexpr int kD = 784;                 // 28*28
constexpr int CH_TILE = 16;             // channels per block (WMMA M dimension)
constexpr int B_TILE  = 8;              // batches per block (one per wave)
constexpr int CHUNKS  = kD / 4;         // 196 float4 chunks per row
constexpr int LDSS    = CHUNKS + 1;     // 197 float4 stride (788 floats) -> 64-bank conflict-free

__global__ __launch_bounds__(256) void chan_linear_wmma(
    const float* __restrict__ x, const float* __restrict__ Wt,
    const float* __restrict__ bias, float* __restrict__ out)
{
  __shared__ float4 wlds[CH_TILE * LDSS];   // 50,432 B

  const int tid = threadIdx.x;
  const int c0  = blockIdx.x * CH_TILE;
  const int b0  = blockIdx.y * B_TILE;

  // ---- stage W[c0 : c0+16, :] into LDS via gfx1250 async global->LDS DMA ----
  for (int t = tid; t < CH_TILE * CHUNKS; t += 256) {
    const int ch = t / CHUNKS;
    const int q  = t - ch * CHUNKS;
    const float4* gsrc = reinterpret_cast<const float4*>(Wt + (size_t)(c0 + ch) * kD) + q;
    float4* ldst = &wlds[ch * LDSS + q];
#if __has_builtin(__builtin_amdgcn_global_load_async_to_lds_b128)
    __builtin_amdgcn_global_load_async_to_lds_b128(
        (g_v4i*)(uintptr_t)gsrc, (l_v4i*)(uintptr_t)ldst, 0, 0);
#else
    *ldst = *gsrc;
#endif
  }
#if __has_builtin(__builtin_amdgcn_global_load_async_to_lds_b128)
#if __has_builtin(__builtin_amdgcn_s_wait_asynccnt)
  __builtin_amdgcn_s_wait_asynccnt(0);
#else
  asm volatile("s_wait_asynccnt 0" ::: "memory");
#endif
#endif
  __syncthreads();

  // ---- streaming dot products ----
  // Lane roles match the ISA f32 16x4 A-matrix layout:
  //   lanes 0-15  : row m = lane,    VGPR pair = {K0,K1} partials
  //   lanes 16-31 : row m = lane-16, VGPR pair = {K2,K3} partials
  const int L = tid & 31;
  const int w = tid >> 5;        // wave -> batch within tile
  const int m = L & 15;          // channel within tile (WMMA row M)
  const int h = L >> 4;          // half-wave -> K pair {0,1} vs {2,3}

  const int b = b0 + w;
  const int c = c0 + m;

  const v4f* __restrict__ xrow =
      reinterpret_cast<const v4f*>(x + ((size_t)b * kC + c) * kD);
  const v4f* wrow = reinterpret_cast<const v4f*>(&wlds[m * LDSS]);

  // Packed accumulators: operands stay register-pair aligned -> pure v_pk_fma_f32.
  v2f acc0 = {0.f, 0.f};
  v2f acc1 = {0.f, 0.f};
#pragma unroll 7
  for (int i = 0; i < 49; ++i) {        // 49 * 4 chunks = 196 = full row
    const int q = 4 * i + 2 * h;
    v4f X0 = xrow[q];
    v4f X1 = xrow[q + 1];
    v4f W0 = wrow[q];
    v4f W1 = wrow[q + 1];
    v2f x0l = __builtin_shufflevector(X0, X0, 0, 1);
    v2f x0h = __builtin_shufflevector(X0, X0, 2, 3);
    v2f x1l = __builtin_shufflevector(X1, X1, 0, 1);
    v2f x1h = __builtin_shufflevector(X1, X1, 2, 3);
    v2f w0l = __builtin_shufflevector(W0, W0, 0, 1);
    v2f w0h = __builtin_shufflevector(W0, W0, 2, 3);
    v2f w1l = __builtin_shufflevector(W1, W1, 0, 1);
    v2f w1h = __builtin_shufflevector(W1, W1, 2, 3);
    acc0 = __builtin_elementwise_fma(x0l, w0l, acc0);
    acc0 = __builtin_elementwise_fma(x0h, w0h, acc0);
    acc1 = __builtin_elementwise_fma(x1l, w1l, acc1);
    acc1 = __builtin_elementwise_fma(x1h, w1h, acc1);
  }

#if __has_builtin(__builtin_amdgcn_wmma_f32_16x16x4_f32)
  // Cross-lane K-reduction on the matrix pipe: B = all-ones (layout-independent),
  // C = 0  =>  D[m,n] = sum_k A[m,k] = full dot product of channel c0+m.
  v2f A = acc0 + acc1;               // one v_pk_add_f32; A is the 2-VGPR fragment
  v2f Bm; Bm.x = 1.f; Bm.y = 1.f;
  v8f Cz = {};
  v8f Dv = __builtin_amdgcn_wmma_f32_16x16x4_f32(
      false, A, false, Bm, (short)0, Cz, false, false);
  // D layout: VGPR r, lanes 0-15 -> D[r, n]; lanes 16-31 -> D[r+8, n] (replicated over n).
  if (m == 0) {                      // lanes 0 (rows 0-7) and 16 (rows 8-15)
    const int cb = c0 + h * 8;
    float* op = out + (size_t)b * kC + cb;
#pragma unroll
    for (int r = 0; r < 8; ++r) op[r] = Dv[r] + bias[cb + r];
  }
#else
  float t = acc0.x + acc0.y + acc1.x + acc1.y;
  t += __shfl_down(t, 16, 32);
  if (h == 0) out[(size_t)b * kC + c] = t + bias[c];
#endif
}

extern "C" void kernel_launch(void* const* d_in, const int* in_sizes, int n_in,
                              void* d_out, int out_size, void* d_ws, size_t ws_size,
                              hipStream_t stream) {
  (void)in_sizes; (void)n_in; (void)d_ws; (void)ws_size; (void)out_size;
  const float* x    = (const float*)d_in[0];
  const float* Wt   = (const float*)d_in[1];
  const float* bias = (const float*)d_in[2];
  float* out        = (float*)d_out;

  dim3 grid(kC / CH_TILE, kB / B_TILE);   // 32 x 32 = 1024 blocks
  chan_linear_wmma<<<grid, dim3(256), 0, stream>>>(x, Wt, bias, out);
}